// WideDeep_7868380086816
// MI455X (gfx1250) — compile-verified
//
#include <hip/hip_runtime.h>
#include <hip/hip_bf16.h>

#define N_NODES 50000
#define F_IN    128
#define H_HEADS 4
#define DH_DIM  64
#define EMB_D   8
#define B_ROWS  16384
#define NS_SP   20
#define V_VOC   1000
#define DENSE_D 64
#define HID_D   256
#define EPS_BN  1e-5f
#define XCAT_D  (NS_SP*EMB_D + DENSE_D)   // 224

typedef float v2f __attribute__((ext_vector_type(2)));
typedef float v8f __attribute__((ext_vector_type(8)));

// ---------- order-preserving float<->uint encoding for atomic max ----------
__device__ __forceinline__ unsigned fenc(float f) {
  unsigned u = __float_as_uint(f);
  return (u & 0x80000000u) ? ~u : (u | 0x80000000u);
}
__device__ __forceinline__ float fdec(unsigned u) {
  return __uint_as_float((u & 0x80000000u) ? (u & 0x7fffffffu) : ~u);
}

// ---------- generic fill ----------
__global__ void fill_u32(unsigned* __restrict__ p, long n, unsigned v) {
  long i = (long)blockIdx.x * blockDim.x + threadIdx.x;
  if (i < n) p[i] = v;
}

// ---------- build x = [sparse_embed | X_dense]  (B x 224) ----------
__global__ void build_xcat(const int* __restrict__ X_sparse,
                           const float* __restrict__ X_dense,
                           const float* __restrict__ emb,
                           float* __restrict__ xcat) {
  int i = blockIdx.x * blockDim.x + threadIdx.x;
  if (i >= B_ROWS * XCAT_D) return;
  int b = i / XCAT_D, c = i % XCAT_D;
  float v;
  if (c < NS_SP * EMB_D) {
    int s = c >> 3, j = c & 7;
    int id = X_sparse[b * NS_SP + s];
    v = emb[((size_t)s * V_VOC + id) * EMB_D + j];
  } else {
    v = X_dense[b * DENSE_D + (c - NS_SP * EMB_D)];
  }
  xcat[i] = v;
}

// ---------- WMMA fp32 GEMM:  C = A(MxK) @ W(NOxK)^T  [+bias][+BN][+relu] ----
// One wave computes one 16x16 output tile via V_WMMA_F32_16X16X4_F32.
__global__ void wmma_gemm_f32(const float* __restrict__ A,
                              const float* __restrict__ W,
                              const float* __restrict__ bias,
                              float* __restrict__ C,
                              int M, int K, int NO,
                              const float* __restrict__ bng,
                              const float* __restrict__ bnb,
                              const float* __restrict__ bnm,
                              const float* __restrict__ bnv,
                              int relu) {
  const int lane = threadIdx.x & 31;
  const int wave = threadIdx.x >> 5;
  const int ntiles = NO >> 4;
  const int tile = blockIdx.x * (blockDim.x >> 5) + wave;
  if (tile >= (M >> 4) * ntiles) return;   // wave-uniform: EXEC stays all-ones
  const int mt = tile / ntiles;
  const int nt = tile % ntiles;
  const int r  = lane & 15;                 // M-row (A) / N-col (B)
  const int kb = (lane >> 4) << 1;          // K sub-offset {0,2} per half-wave
  const float* ap = A + (size_t)(mt * 16 + r) * K + kb;
  const float* bp = W + (size_t)(nt * 16 + r) * K + kb;
  v8f acc = {};
  for (int k0 = 0; k0 < K; k0 += 4) {
    v2f a = *(const v2f*)(ap + k0);
    v2f b = *(const v2f*)(bp + k0);
    acc = __builtin_amdgcn_wmma_f32_16x16x4_f32(false, a, false, b,
                                                (short)0, acc, false, false);
  }
  const int ncol = nt * 16 + r;
  float bs = bias ? bias[ncol] : 0.0f;
  float g = 1.f, bb = 0.f, mm = 0.f, inv = 1.f;
  if (bng) {
    g = bng[ncol]; bb = bnb[ncol]; mm = bnm[ncol];
    inv = 1.0f / sqrtf(bnv[ncol] + EPS_BN);
  }
  const int mbase = mt * 16 + ((lane >> 4) << 3);  // rows r..r+7 / +8
#pragma unroll
  for (int i = 0; i < 8; i++) {
    float v = acc[i] + bs;
    if (bng) v = g * (v - mm) * inv + bb;
    if (relu) v = v > 0.f ? v : 0.f;
    C[(size_t)(mbase + i) * NO + ncol] = v;
  }
}

// ---------- small dense GEMM (narrow outputs, VALU) ----------
__global__ void small_gemm(const float* __restrict__ A, const float* __restrict__ W,
                           const float* __restrict__ bias, float* __restrict__ C,
                           int M, int K, int O) {
  int i = blockIdx.x * blockDim.x + threadIdx.x;
  if (i >= M * O) return;
  int m = i / O, o = i % O;
  const float* a = A + (size_t)m * K;
  const float* w = W + (size_t)o * K;
  float s = bias ? bias[o] : 0.f;
  for (int k = 0; k < K; k++) s += a[k] * w[k];
  C[i] = s;
}

// ---------- per-node attention logits el/er ----------
__global__ void el_er_kernel(const float* __restrict__ feat,
                             const float* __restrict__ al, const float* __restrict__ ar,
                             float* __restrict__ el, float* __restrict__ er,
                             int H, int D) {
  int i = blockIdx.x * blockDim.x + threadIdx.x;
  if (i >= N_NODES * H) return;
  int n = i / H, h = i % H;
  const float* f  = feat + (size_t)n * H * D + h * D;
  const float* pl = al + h * D;
  const float* pr = ar + h * D;
  float sl = 0.f, sr = 0.f;
  for (int d = 0; d < D; d++) { sl += f[d] * pl[d]; sr += f[d] * pr[d]; }
  el[i] = sl; er[i] = sr;
}

// ---------- edge pass 1: segment max of leaky_relu(el[src]+er[dst]) ----------
__global__ void edge_max(const int* __restrict__ src, const int* __restrict__ dst,
                         const float* __restrict__ el, const float* __restrict__ er,
                         unsigned* __restrict__ emax, int Etot, int H) {
  int i = blockIdx.x * blockDim.x + threadIdx.x;
  if (i >= Etot * H) return;
  int e = i / H, h = i % H;
  float x = el[src[e] * H + h] + er[dst[e] * H + h];
  x = x > 0.f ? x : 0.2f * x;
  atomicMax(&emax[dst[e] * H + h], fenc(x));
}

// ---------- edge pass 2: a = exp(e - emax[dst]);  denom[dst] += a ----------
__global__ void edge_exp(const int* __restrict__ src, const int* __restrict__ dst,
                         const float* __restrict__ el, const float* __restrict__ er,
                         const unsigned* __restrict__ emax,
                         float* __restrict__ a_e, float* __restrict__ denom,
                         int Etot, int H) {
  int i = blockIdx.x * blockDim.x + threadIdx.x;
  if (i >= Etot * H) return;
  int e = i / H, h = i % H;
  int d = dst[e];
  float x = el[src[e] * H + h] + er[d * H + h];
  x = x > 0.f ? x : 0.2f * x;
  float a = expf(x - fdec(emax[d * H + h]));
  a_e[i] = a;
  atomicAdd(&denom[d * H + h], a);
}

// ---------- edge pass 3 (layer1): agg[dst] += a * feat[src]  (H=4, DH=64) ----
__global__ void edge_agg1(const int* __restrict__ src, const int* __restrict__ dst,
                          const float* __restrict__ a_e, const float* __restrict__ feat,
                          float* __restrict__ agg, int Etot) {
  int i = blockIdx.x * blockDim.x + threadIdx.x;
  if (i >= Etot * H_HEADS * 16) return;          // 16 chunks of 4 floats per head
  int c4 = i & 15;
  int t  = i >> 4;
  int h  = t & 3;
  int e  = t >> 2;
  int s = src[e], d = dst[e];
  float a = a_e[e * H_HEADS + h];
  float4 f = *(const float4*)(feat + (size_t)s * (H_HEADS * DH_DIM) + h * DH_DIM + c4 * 4);
  float* q = agg + (size_t)d * (H_HEADS * DH_DIM) + h * DH_DIM + c4 * 4;
  atomicAdd(q + 0, a * f.x);
  atomicAdd(q + 1, a * f.y);
  atomicAdd(q + 2, a * f.z);
  atomicAdd(q + 3, a * f.w);
}

// ---------- edge pass 3 (layer2): H=1, D=8 ----------
__global__ void edge_agg2(const int* __restrict__ src, const int* __restrict__ dst,
                          const float* __restrict__ a_e, const float* __restrict__ feat,
                          float* __restrict__ agg, int Etot) {
  int i = blockIdx.x * blockDim.x + threadIdx.x;
  if (i >= Etot * EMB_D) return;
  int e = i >> 3, j = i & 7;
  atomicAdd(&agg[(size_t)dst[e] * EMB_D + j],
            a_e[e] * feat[(size_t)src[e] * EMB_D + j]);
}

// ---------- finalize layer 1: (agg/denom + bias) -> ELU, in place ----------
__global__ void finalize1(float* __restrict__ agg, const float* __restrict__ denom,
                          const float* __restrict__ bias) {
  int i = blockIdx.x * blockDim.x + threadIdx.x;
  if (i >= N_NODES * HID_D) return;
  int n = i >> 8, c = i & 255, h = c >> 6;
  float v = agg[i] / denom[n * H_HEADS + h] + bias[c];
  agg[i] = v > 0.f ? v : expf(v) - 1.f;
}

// ---------- finalize layer 2: agg/denom + bias (no act), in place ----------
__global__ void finalize2(float* __restrict__ agg, const float* __restrict__ denom,
                          const float* __restrict__ bias) {
  int i = blockIdx.x * blockDim.x + threadIdx.x;
  if (i >= N_NODES * EMB_D) return;
  int n = i >> 3, j = i & 7;
  agg[i] = agg[i] / denom[n] + bias[j];
}

// ---------- fusion head: gates + combine + gout + softmax ----------
__global__ void final_head(const int* __restrict__ X_id,
                           const float* __restrict__ wide_o,
                           const float* __restrict__ deep_o,
                           const float* __restrict__ gcn,
                           const float* __restrict__ g1w, const float* __restrict__ g1b,
                           const float* __restrict__ g2w, const float* __restrict__ g2b,
                           const float* __restrict__ g3w, const float* __restrict__ g3b,
                           const float* __restrict__ gout_w, const float* __restrict__ gout_b,
                           float* __restrict__ out) {
  int b = blockIdx.x * blockDim.x + threadIdx.x;
  if (b >= B_ROWS) return;
  int commit = X_id[b * 2 + 1];
  const float* w = wide_o + (size_t)b * EMB_D;
  const float* d = deep_o + (size_t)b * EMB_D;
  const float* g = gcn + (size_t)commit * EMB_D;
  float sa = g1b[0], sb = g2b[0], sc = g3b[0];
#pragma unroll
  for (int j = 0; j < 8; j++) { sa += w[j] * g1w[j]; sb += d[j] * g2w[j]; sc += g[j] * g3w[j]; }
  float aw = 1.f / (1.f + expf(-sa));
  float bw = 1.f / (1.f + expf(-sb));
  float cw = 1.f / (1.f + expf(-sc));
  float tot = aw + bw + cw;
  float comb[24];
#pragma unroll
  for (int j = 0; j < 8; j++) {
    comb[j]      = aw / tot * w[j];
    comb[8 + j]  = bw / tot * d[j];
    comb[16 + j] = cw / tot * g[j];
  }
  float o0 = gout_b[0], o1 = gout_b[1];
#pragma unroll
  for (int j = 0; j < 24; j++) { o0 += comb[j] * gout_w[j]; o1 += comb[j] * gout_w[24 + j]; }
  float mx = fmaxf(o0, o1);
  float e0 = expf(o0 - mx), e1 = expf(o1 - mx);
  float s = e0 + e1;
  out[b * 2 + 0] = e0 / s;
  out[b * 2 + 1] = e1 / s;
}

static inline int gridFor(long n, int bs) { return (int)((n + bs - 1) / bs); }

extern "C" void kernel_launch(void* const* d_in, const int* in_sizes, int n_in,
                              void* d_out, int out_size, void* d_ws, size_t ws_size,
                              hipStream_t stream) {
  // ---- inputs (setup_inputs dict order) ----
  const int*   X_id      = (const int*)  d_in[0];
  const int*   X_sparse  = (const int*)  d_in[1];
  const float* X_dense   = (const float*)d_in[2];
  const float* emb       = (const float*)d_in[3];
  const float* gfeat     = (const float*)d_in[4];
  const int*   src       = (const int*)  d_in[5];
  const int*   dst       = (const int*)  d_in[6];
  const float* wide_w    = (const float*)d_in[7];
  const float* wide_b    = (const float*)d_in[8];
  const float* lin1_w    = (const float*)d_in[9];
  const float* lin1_b    = (const float*)d_in[10];
  const float* bn1_g     = (const float*)d_in[11];
  const float* bn1_b     = (const float*)d_in[12];
  const float* bn1_m     = (const float*)d_in[13];
  const float* bn1_v     = (const float*)d_in[14];
  const float* lin2_w    = (const float*)d_in[15];
  const float* lin2_b    = (const float*)d_in[16];
  const float* bn2_g     = (const float*)d_in[17];
  const float* bn2_b     = (const float*)d_in[18];
  const float* bn2_m     = (const float*)d_in[19];
  const float* bn2_v     = (const float*)d_in[20];
  const float* dnn_w     = (const float*)d_in[21];
  const float* dnn_b     = (const float*)d_in[22];
  const float* gat1_w    = (const float*)d_in[23];
  const float* gat1_al   = (const float*)d_in[24];
  const float* gat1_ar   = (const float*)d_in[25];
  const float* gat1_bias = (const float*)d_in[26];
  const float* gat2_w    = (const float*)d_in[27];
  const float* gat2_al   = (const float*)d_in[28];
  const float* gat2_ar   = (const float*)d_in[29];
  const float* gat2_bias = (const float*)d_in[30];
  const float* g1w = (const float*)d_in[31];
  const float* g1b = (const float*)d_in[32];
  const float* g2w = (const float*)d_in[33];
  const float* g2b = (const float*)d_in[34];
  const float* g3w = (const float*)d_in[35];
  const float* g3b = (const float*)d_in[36];
  const float* gout_w = (const float*)d_in[37];
  const float* gout_b = (const float*)d_in[38];
  const int Etot = in_sizes[5];   // E + N = 450000

  // ---- workspace carve (256B aligned) ----
  char* ws = (char*)d_ws;
  size_t off = 0;
  auto carve = [&](size_t bytes) -> void* {
    void* p = ws + off;
    off = (off + bytes + 255) & ~(size_t)255;
    return p;
  };
  float*    feat1  = (float*)   carve((size_t)N_NODES * HID_D * 4);   // gat1 transform; later feat2
  float*    agg1   = (float*)   carve((size_t)N_NODES * HID_D * 4);   // gat1 aggregate -> h1 (in place)
  float*    el1    = (float*)   carve((size_t)N_NODES * H_HEADS * 4);
  float*    er1    = (float*)   carve((size_t)N_NODES * H_HEADS * 4);
  unsigned* emax1  = (unsigned*)carve((size_t)N_NODES * H_HEADS * 4);
  float*    denom1 = (float*)   carve((size_t)N_NODES * H_HEADS * 4);
  float*    a_e    = (float*)   carve((size_t)Etot * H_HEADS * 4);    // per-edge exp; reused layer2
  float*    agg2   = (float*)   carve((size_t)N_NODES * EMB_D * 4);   // -> gcn (in place)
  float*    el2    = (float*)   carve((size_t)N_NODES * 4);
  float*    er2    = (float*)   carve((size_t)N_NODES * 4);
  unsigned* emax2  = (unsigned*)carve((size_t)N_NODES * 4);
  float*    denom2 = (float*)   carve((size_t)N_NODES * 4);
  float*    xcat   = (float*)   carve((size_t)B_ROWS * XCAT_D * 4);
  float*    dd1    = (float*)   carve((size_t)B_ROWS * HID_D * 4);
  float*    dd2    = (float*)   carve((size_t)B_ROWS * HID_D * 4);
  float*    wide_o = (float*)   carve((size_t)B_ROWS * EMB_D * 4);
  float*    deep_o = (float*)   carve((size_t)B_ROWS * EMB_D * 4);
  (void)ws_size; (void)n_in; (void)out_size;

  const int TB = 256;

  // ---- zero accumulators (deterministic every call) ----
  fill_u32<<<gridFor((long)N_NODES * HID_D, TB), TB, 0, stream>>>((unsigned*)agg1,  (long)N_NODES * HID_D, 0u);
  fill_u32<<<gridFor((long)N_NODES * H_HEADS, TB), TB, 0, stream>>>(emax1,          (long)N_NODES * H_HEADS, 0u);
  fill_u32<<<gridFor((long)N_NODES * H_HEADS, TB), TB, 0, stream>>>((unsigned*)denom1,(long)N_NODES * H_HEADS, 0u);
  fill_u32<<<gridFor((long)N_NODES * EMB_D, TB), TB, 0, stream>>>((unsigned*)agg2,  (long)N_NODES * EMB_D, 0u);
  fill_u32<<<gridFor((long)N_NODES, TB), TB, 0, stream>>>(emax2,                    (long)N_NODES, 0u);
  fill_u32<<<gridFor((long)N_NODES, TB), TB, 0, stream>>>((unsigned*)denom2,        (long)N_NODES, 0u);

  // ---- deep path ----
  build_xcat<<<gridFor((long)B_ROWS * XCAT_D, TB), TB, 0, stream>>>(X_sparse, X_dense, emb, xcat);
  // lin1 + BN1 + ReLU  (WMMA)
  wmma_gemm_f32<<<(B_ROWS / 16) * (HID_D / 16) / 4, 128, 0, stream>>>(
      xcat, lin1_w, lin1_b, dd1, B_ROWS, XCAT_D, HID_D, bn1_g, bn1_b, bn1_m, bn1_v, 1);
  // lin2 + BN2 + ReLU  (WMMA)
  wmma_gemm_f32<<<(B_ROWS / 16) * (HID_D / 16) / 4, 128, 0, stream>>>(
      dd1, lin2_w, lin2_b, dd2, B_ROWS, HID_D, HID_D, bn2_g, bn2_b, bn2_m, bn2_v, 1);
  // deep_out / wide_out (narrow)
  small_gemm<<<gridFor((long)B_ROWS * EMB_D, TB), TB, 0, stream>>>(dd2, dnn_w, dnn_b, deep_o, B_ROWS, HID_D, EMB_D);
  small_gemm<<<gridFor((long)B_ROWS * EMB_D, TB), TB, 0, stream>>>(X_dense, wide_w, wide_b, wide_o, B_ROWS, DENSE_D, EMB_D);

  // ---- GAT layer 1 ----
  wmma_gemm_f32<<<(N_NODES / 16) * (HID_D / 16) / 4, 128, 0, stream>>>(
      gfeat, gat1_w, nullptr, feat1, N_NODES, F_IN, HID_D, nullptr, nullptr, nullptr, nullptr, 0);
  el_er_kernel<<<gridFor((long)N_NODES * H_HEADS, TB), TB, 0, stream>>>(feat1, gat1_al, gat1_ar, el1, er1, H_HEADS, DH_DIM);
  edge_max<<<gridFor((long)Etot * H_HEADS, TB), TB, 0, stream>>>(src, dst, el1, er1, emax1, Etot, H_HEADS);
  edge_exp<<<gridFor((long)Etot * H_HEADS, TB), TB, 0, stream>>>(src, dst, el1, er1, emax1, a_e, denom1, Etot, H_HEADS);
  edge_agg1<<<gridFor((long)Etot * H_HEADS * 16, TB), TB, 0, stream>>>(src, dst, a_e, feat1, agg1, Etot);
  finalize1<<<gridFor((long)N_NODES * HID_D, TB), TB, 0, stream>>>(agg1, denom1, gat1_bias);   // agg1 -> h1

  // ---- GAT layer 2 (feat2 overwrites feat1 region; h1 lives in agg1) ----
  float* feat2 = feat1;
  small_gemm<<<gridFor((long)N_NODES * EMB_D, TB), TB, 0, stream>>>(agg1, gat2_w, nullptr, feat2, N_NODES, HID_D, EMB_D);
  el_er_kernel<<<gridFor((long)N_NODES, TB), TB, 0, stream>>>(feat2, gat2_al, gat2_ar, el2, er2, 1, EMB_D);
  edge_max<<<gridFor((long)Etot, TB), TB, 0, stream>>>(src, dst, el2, er2, emax2, Etot, 1);
  edge_exp<<<gridFor((long)Etot, TB), TB, 0, stream>>>(src, dst, el2, er2, emax2, a_e, denom2, Etot, 1);
  edge_agg2<<<gridFor((long)Etot * EMB_D, TB), TB, 0, stream>>>(src, dst, a_e, feat2, agg2, Etot);
  finalize2<<<gridFor((long)N_NODES * EMB_D, TB), TB, 0, stream>>>(agg2, denom2, gat2_bias);   // agg2 -> gcn

  // ---- fusion head ----
  final_head<<<gridFor((long)B_ROWS, TB), TB, 0, stream>>>(
      X_id, wide_o, deep_o, agg2, g1w, g1b, g2w, g2b, g3w, g3b, gout_w, gout_b, (float*)d_out);
}